// GraphSage_69217692942518
// MI455X (gfx1250) — compile-verified
//
#include <hip/hip_runtime.h>

typedef float v2f __attribute__((ext_vector_type(2)));
typedef float v8f __attribute__((ext_vector_type(8)));

#define N_NODES 100000
#define N_EDGES 3200000
#define D_IN    128
#define D_HID   128
#define D_OUT   64

// ---------------------------------------------------------------------------
// Zero-fill (float4 vectorized). Workspace is poisoned 0xAA by the harness,
// so the accumulator must be cleared explicitly every call.
// ---------------------------------------------------------------------------
__global__ void zero4_kernel(float4* __restrict__ p, size_t n4) {
    size_t i      = (size_t)blockIdx.x * blockDim.x + threadIdx.x;
    size_t stride = (size_t)gridDim.x * blockDim.x;
    float4 z = make_float4(0.f, 0.f, 0.f, 0.f);
    for (; i < n4; i += stride) p[i] = z;
}

// ---------------------------------------------------------------------------
// Edge scatter: one wave (32 lanes) per edge, each lane owns a float4 chunk
// of the 128-wide feature row. agg[dst] += feat[src] * w via f32 atomics.
// x/h (51.2MB) + agg (51.2MB) both fit in the 192MB L2, so the random
// gather + atomic RMW traffic stays L2-resident on MI455X.
// ---------------------------------------------------------------------------
__global__ void scatter_kernel(const float* __restrict__ feat,
                               const int*   __restrict__ src,
                               const int*   __restrict__ dst,
                               const float* __restrict__ ew,
                               float*       __restrict__ agg,
                               float*       __restrict__ cnt,
                               int nE, int do_cnt) {
    size_t total  = (size_t)nE * 32;                  // 32 lanes per edge
    size_t i      = (size_t)blockIdx.x * blockDim.x + threadIdx.x;
    size_t stride = (size_t)gridDim.x * blockDim.x;
    for (; i < total; i += stride) {
        int e = (int)(i >> 5);
        int q = (int)(i & 31);                        // which float4 of the row
        int s = src[e];
        int d = dst[e];
        float w = ew[e];
        const float4 v = ((const float4*)(feat + (size_t)s * 128))[q];
        float* dp = agg + (size_t)d * 128 + (size_t)q * 4;
        atomicAdd(dp + 0, v.x * w);
        atomicAdd(dp + 1, v.y * w);
        atomicAdd(dp + 2, v.z * w);
        atomicAdd(dp + 3, v.w * w);
        if (do_cnt && q == 0) atomicAdd(cnt + d, 1.0f);
    }
}

// ---------------------------------------------------------------------------
// scale[n] = 1 / max(cnt[n], 1)  (segment-mean normalization, hoisted so the
// GEMM can fold it into the A-tile load)
// ---------------------------------------------------------------------------
__global__ void scale_kernel(const float* __restrict__ cnt,
                             float* __restrict__ rscale, int n) {
    int i = blockIdx.x * blockDim.x + threadIdx.x;
    if (i < n) rscale[i] = 1.0f / fmaxf(cnt[i], 1.0f);
}

// ---------------------------------------------------------------------------
// Fused SAGE dense layer via V_WMMA_F32_16X16X4_F32 (wave32, f32-exact):
//   out[m, n] = act( sum_k agg[m,k]*scale[m]*Wl[n,k] + sum_k root[m,k]*Wr[n,k] + b[n] )
// One wave per 16x16 output tile, 8 waves per 256-thread block.
// A-tile layout (ISA 7.12.2, 32-bit A 16x4): lane m=lane&15 holds K = (lane>>4)*2 .. +1
// B-tile mirrored; C/D: lane's column n = lane&15, rows v + (lane>=16 ? 8 : 0).
// 100000 rows = 6250 * 16 exactly -> no remainder tiles, EXEC all-ones.
// ---------------------------------------------------------------------------
__global__ void sage_gemm_kernel(const float* __restrict__ Aagg,
                                 const float* __restrict__ Aroot,
                                 const float* __restrict__ Wl,
                                 const float* __restrict__ Wr,
                                 const float* __restrict__ bias,
                                 const float* __restrict__ rscale,
                                 float*       __restrict__ out,
                                 int NT, int Nout, int do_relu) {
    const int K = 128;
    int wave = threadIdx.x >> 5;
    int lane = threadIdx.x & 31;
    int tile = blockIdx.x * 8 + wave;          // 8 waves per block
    int row_tile = tile / NT;
    int n_tile   = tile % NT;

    int m    = lane & 15;
    int hi   = lane >> 4;                      // 0: K=0,1   1: K=2,3
    int koff = hi * 2;

    int row = row_tile * 16 + m;               // A-matrix row for this lane
    int col = n_tile * 16 + m;                 // B-matrix column (= output channel)

    const float* ar = Aagg  + (size_t)row * K;
    const float* xr = Aroot + (size_t)row * K;
    const float* wl = Wl    + (size_t)col * K; // W[o][k] row-major -> B[k][n]=W[n][k]
    const float* wr = Wr    + (size_t)col * K;
    float s = rscale[row];

    v8f acc1 = {0.f, 0.f, 0.f, 0.f, 0.f, 0.f, 0.f, 0.f};
    v8f acc2 = {0.f, 0.f, 0.f, 0.f, 0.f, 0.f, 0.f, 0.f};

    // agg * scale  @  Wl^T
    #pragma unroll
    for (int k = 0; k < K; k += 4) {
        v2f a = *(const v2f*)(ar + k + koff);
        a *= s;
        v2f b = *(const v2f*)(wl + k + koff);
        acc1 = __builtin_amdgcn_wmma_f32_16x16x4_f32(
            false, a, false, b, (short)0, acc1, false, false);
    }
    // root @ Wr^T  (independent accumulation chain -> overlaps with chain 1)
    #pragma unroll
    for (int k = 0; k < K; k += 4) {
        v2f a = *(const v2f*)(xr + k + koff);
        v2f b = *(const v2f*)(wr + k + koff);
        acc2 = __builtin_amdgcn_wmma_f32_16x16x4_f32(
            false, a, false, b, (short)0, acc2, false, false);
    }

    v8f acc = acc1 + acc2;
    float bv = bias[col];                      // lane's column == store column
    int mbase = row_tile * 16 + hi * 8;
    int ncol  = n_tile * 16 + (lane & 15);
    #pragma unroll
    for (int v = 0; v < 8; ++v) {
        float val = acc[v] + bv;
        if (do_relu) val = fmaxf(val, 0.0f);
        out[(size_t)(mbase + v) * Nout + ncol] = val;
    }
}

// ---------------------------------------------------------------------------
extern "C" void kernel_launch(void* const* d_in, const int* in_sizes, int n_in,
                              void* d_out, int out_size, void* d_ws, size_t ws_size,
                              hipStream_t stream) {
    (void)in_sizes; (void)n_in; (void)out_size; (void)ws_size;

    const float* x   = (const float*)d_in[0];
    const int*   ei  = (const int*)  d_in[1];   // [2, N_EDGES] (JAX default int32)
    const float* ew  = (const float*)d_in[2];
    const float* Wl1 = (const float*)d_in[3];
    const float* bl1 = (const float*)d_in[4];
    const float* Wr1 = (const float*)d_in[5];
    const float* Wl2 = (const float*)d_in[6];
    const float* bl2 = (const float*)d_in[7];
    const float* Wr2 = (const float*)d_in[8];

    const int* src = ei;
    const int* dst = ei + N_EDGES;

    float* out    = (float*)d_out;
    float* h      = out;                                   // [N_NODES, 128]
    float* logits = out + (size_t)N_NODES * D_HID;         // [N_NODES, 64]

    // workspace layout: agg [N_NODES*128] | cnt [N_NODES] | rscale [N_NODES]
    float* agg    = (float*)d_ws;
    float* cnt    = agg + (size_t)N_NODES * 128;
    float* rscale = cnt + N_NODES;

    // ---- layer 1 ----
    size_t zn4 = ((size_t)N_NODES * 128 + N_NODES) / 4;    // agg + cnt (contiguous)
    zero4_kernel<<<4096, 256, 0, stream>>>((float4*)agg, zn4);
    scatter_kernel<<<8192, 256, 0, stream>>>(x, src, dst, ew, agg, cnt, N_EDGES, 1);
    scale_kernel<<<(N_NODES + 255) / 256, 256, 0, stream>>>(cnt, rscale, N_NODES);
    // 6250 row tiles * 8 n-tiles / 8 waves-per-block = 6250 blocks
    sage_gemm_kernel<<<6250, 256, 0, stream>>>(agg, x, Wl1, Wr1, bl1, rscale,
                                               h, /*NT=*/8, /*Nout=*/D_HID, /*relu=*/1);

    // ---- layer 2 (reuse agg buffer and degree counts) ----
    zero4_kernel<<<4096, 256, 0, stream>>>((float4*)agg, ((size_t)N_NODES * 128) / 4);
    scatter_kernel<<<8192, 256, 0, stream>>>(h, src, dst, ew, agg, cnt, N_EDGES, 0);
    // 6250 row tiles * 4 n-tiles / 8 = 3125 blocks
    sage_gemm_kernel<<<3125, 256, 0, stream>>>(agg, h, Wl2, Wr2, bl2, rscale,
                                               logits, /*NT=*/4, /*Nout=*/D_OUT, /*relu=*/0);
}